// PYG_ChebNet_15118284881953
// MI455X (gfx1250) — compile-verified
//
#include <hip/hip_runtime.h>
#include <hip/hip_bf16.h>
#include <math.h>

#define N_NODES 100000
#define N_EDGES 1600000
#define IN_C 128
#define OUT_C 64
#define NUM_CLASS 40

typedef __attribute__((ext_vector_type(2))) float v2f;
typedef __attribute__((ext_vector_type(8))) float v8f;

// ---------------- elementwise / edge kernels ----------------

__global__ __launch_bounds__(256) void k_zero(float* __restrict__ p, int n) {
  int i = blockIdx.x * blockDim.x + threadIdx.x;
  if (i < n) p[i] = 0.0f;
}

__global__ __launch_bounds__(256) void k_deg(const long long* __restrict__ ei,
                                             const float* __restrict__ w,
                                             float* __restrict__ deg) {
  int e = blockIdx.x * blockDim.x + threadIdx.x;
  if (e < N_EDGES) {
    int s = (int)ei[e];  // row of edge_index = src
    unsafeAtomicAdd(&deg[s], w[e]);
  }
}

__global__ __launch_bounds__(256) void k_dinv(float* __restrict__ deg) {
  int i = blockIdx.x * blockDim.x + threadIdx.x;
  if (i < N_NODES) {
    float d = deg[i];
    deg[i] = (d > 0.0f) ? rsqrtf(d) : 0.0f;  // in-place -> dinv
  }
}

__global__ __launch_bounds__(256) void k_norm(const long long* __restrict__ ei,
                                              const float* __restrict__ w,
                                              const float* __restrict__ dinv,
                                              float* __restrict__ norm) {
  int e = blockIdx.x * blockDim.x + threadIdx.x;
  if (e < N_EDGES) {
    int s = (int)ei[e];
    int d = (int)ei[(long long)N_EDGES + e];
    // (2/LAMBDA_MAX) == 1.0
    norm[e] = -dinv[s] * w[e] * dinv[d];
  }
}

// one wave32 per edge: lane covers 2 of 64 channels (float2), atomic scatter to dst
__global__ __launch_bounds__(256) void k_prop(const long long* __restrict__ ei,
                                              const float* __restrict__ norm,
                                              const float* __restrict__ V,
                                              float* __restrict__ P) {
  const int wave = threadIdx.x >> 5;
  const int lane = threadIdx.x & 31;
  const int e = blockIdx.x * 8 + wave;
  if (e >= N_EDGES) return;
  const int s = (int)ei[e];
  const int d = (int)ei[(long long)N_EDGES + e];
  const float w = norm[e];
  const float2 v = *(const float2*)(V + (size_t)s * OUT_C + lane * 2);
  float* p = P + (size_t)d * OUT_C + lane * 2;
  unsafeAtomicAdd(p, w * v.x);
  unsafeAtomicAdd(p + 1, w * v.y);
}

// T2 = 2*prop(T1) - T0   (prop result already in T2)
__global__ __launch_bounds__(256) void k_t2(float* __restrict__ T2,
                                            const float* __restrict__ T0, int n) {
  int i = blockIdx.x * blockDim.x + threadIdx.x;
  if (i < n) T2[i] = 2.0f * T2[i] - T0[i];
}

// ---------------- WMMA fp32 GEMM kernels (16x16 tile per wave) ----------------
// A operand (16x4 f32): lanes 0-15 hold K=0(V0),1(V1); lanes 16-31 hold K=2,3
// B operand (4x16 f32): V0 = row K=0 (lanes0-15) / K=2 (lanes16-31); V1 = K=1 / K=3
// C/D (16x16 f32): VGPR r -> M=r (lanes0-15), M=r+8 (lanes16-31); N = lane&15

// H = tanh(X @ W_in + b_in)   [100000,128] x [128,64]
__global__ __launch_bounds__(256) void k_gemm_in(const float* __restrict__ X,
                                                 const float* __restrict__ W,
                                                 const float* __restrict__ bias,
                                                 float* __restrict__ H) {
  const int wave = threadIdx.x >> 5;
  const int lane = threadIdx.x & 31;
  const int tile = blockIdx.x * 8 + wave;  // 25000 tiles, grid is exact
  const int tm = tile >> 2;                // 6250 row tiles
  const int tn = tile & 3;                 // 4 col tiles
  const int half = lane >> 4;
  const int l15 = lane & 15;
  const int rowA = tm * 16 + l15;
  const int colB = tn * 16 + l15;
  const float* A = X + (size_t)rowA * IN_C + 2 * half;
  const float* B = W + (size_t)(2 * half) * OUT_C + colB;
  v8f c = {};
#pragma unroll
  for (int k0 = 0; k0 < IN_C; k0 += 4) {
    v2f a, b;
    a.x = A[k0];            a.y = A[k0 + 1];
    b.x = B[k0 * OUT_C];    b.y = B[(k0 + 1) * OUT_C];
    c = __builtin_amdgcn_wmma_f32_16x16x4_f32(false, a, false, b, (short)0, c, false, false);
  }
  const float bc = bias[colB];
#pragma unroll
  for (int r = 0; r < 8; ++r) {
    int m = tm * 16 + r + 8 * half;
    H[(size_t)m * OUT_C + colB] = tanhf(c[r] + bc);
  }
}

// ACC = relu(T0@th0 + T1@th1 + T2@th2 + cheb_bias)   K=64 each
__global__ __launch_bounds__(256) void k_gemm_cheb(const float* __restrict__ T0,
                                                   const float* __restrict__ T1,
                                                   const float* __restrict__ T2,
                                                   const float* __restrict__ theta,
                                                   const float* __restrict__ cb,
                                                   float* __restrict__ ACC) {
  const int wave = threadIdx.x >> 5;
  const int lane = threadIdx.x & 31;
  const int tile = blockIdx.x * 8 + wave;
  const int tm = tile >> 2;
  const int tn = tile & 3;
  const int half = lane >> 4;
  const int l15 = lane & 15;
  const int rowA = tm * 16 + l15;
  const int colB = tn * 16 + l15;
  v8f c = {};
  const float* Ts[3] = {T0, T1, T2};
#pragma unroll
  for (int k = 0; k < 3; ++k) {
    const float* A = Ts[k] + (size_t)rowA * OUT_C + 2 * half;
    const float* B = theta + (size_t)k * OUT_C * OUT_C + (size_t)(2 * half) * OUT_C + colB;
#pragma unroll
    for (int k0 = 0; k0 < OUT_C; k0 += 4) {
      v2f a, b;
      a.x = A[k0];            a.y = A[k0 + 1];
      b.x = B[k0 * OUT_C];    b.y = B[(k0 + 1) * OUT_C];
      c = __builtin_amdgcn_wmma_f32_16x16x4_f32(false, a, false, b, (short)0, c, false, false);
    }
  }
  const float bc = cb[colB];
#pragma unroll
  for (int r = 0; r < 8; ++r) {
    int m = tm * 16 + r + 8 * half;
    ACC[(size_t)m * OUT_C + colB] = fmaxf(c[r] + bc, 0.0f);
  }
}

// Y = ACC @ W_out + b_out   [100000,64] x [64,40]; 3 col-tiles, mask cols >= 40
__global__ __launch_bounds__(256) void k_gemm_out(const float* __restrict__ ACC,
                                                  const float* __restrict__ W,
                                                  const float* __restrict__ bias,
                                                  float* __restrict__ Y) {
  const int wave = threadIdx.x >> 5;
  const int lane = threadIdx.x & 31;
  const int tile = blockIdx.x * 8 + wave;
  const int NT = (N_NODES / 16) * 3;  // 18750
  if (tile >= NT) return;             // wave-uniform exit (EXEC stays full for WMMA)
  const int tm = tile / 3;
  const int tn = tile % 3;
  const int half = lane >> 4;
  const int l15 = lane & 15;
  const int rowA = tm * 16 + l15;
  const int colB = tn * 16 + l15;
  const int colC = colB < NUM_CLASS ? colB : (NUM_CLASS - 1);  // clamp (no divergent loads)
  const float cmask = colB < NUM_CLASS ? 1.0f : 0.0f;
  const float* A = ACC + (size_t)rowA * OUT_C + 2 * half;
  const float* B = W + (size_t)(2 * half) * NUM_CLASS + colC;
  v8f c = {};
#pragma unroll
  for (int k0 = 0; k0 < OUT_C; k0 += 4) {
    v2f a, b;
    a.x = A[k0];                      a.y = A[k0 + 1];
    b.x = cmask * B[k0 * NUM_CLASS];  b.y = cmask * B[(k0 + 1) * NUM_CLASS];
    c = __builtin_amdgcn_wmma_f32_16x16x4_f32(false, a, false, b, (short)0, c, false, false);
  }
  if (colB < NUM_CLASS) {
    const float bc = bias[colB];
#pragma unroll
    for (int r = 0; r < 8; ++r) {
      int m = tm * 16 + r + 8 * half;
      Y[(size_t)m * NUM_CLASS + colB] = c[r] + bc;
    }
  }
}

// ---------------- driver ----------------

extern "C" void kernel_launch(void* const* d_in, const int* in_sizes, int n_in,
                              void* d_out, int out_size, void* d_ws, size_t ws_size,
                              hipStream_t stream) {
  const float*     x     = (const float*)d_in[0];
  const long long* ei    = (const long long*)d_in[1];  // int64 [2,E]
  const float*     ew    = (const float*)d_in[2];
  const float*     W_in  = (const float*)d_in[3];
  const float*     b_in  = (const float*)d_in[4];
  const float*     theta = (const float*)d_in[5];
  const float*     cb    = (const float*)d_in[6];
  const float*     W_out = (const float*)d_in[7];
  const float*     b_out = (const float*)d_in[8];
  float* Y = (float*)d_out;

  float* ws = (float*)d_ws;
  size_t o = 0;
  float* deg  = ws + o; o += ((N_NODES + 1023) / 1024) * 1024;
  float* norm = ws + o; o += (((size_t)N_EDGES + 1023) / 1024) * 1024;
  float* T0   = ws + o; o += (size_t)N_NODES * OUT_C;
  float* T1   = ws + o; o += (size_t)N_NODES * OUT_C;
  float* T2   = ws + o; o += (size_t)N_NODES * OUT_C;
  float* ACC  = ws + o; o += (size_t)N_NODES * OUT_C;   // total ~109 MB

  const int NF = N_NODES * OUT_C;  // 6.4M

  // normalization coefficients
  k_zero<<<(N_NODES + 255) / 256, 256, 0, stream>>>(deg, N_NODES);
  k_zero<<<(NF + 255) / 256, 256, 0, stream>>>(T1, NF);
  k_zero<<<(NF + 255) / 256, 256, 0, stream>>>(T2, NF);
  k_deg<<<(N_EDGES + 255) / 256, 256, 0, stream>>>(ei, ew, deg);
  k_dinv<<<(N_NODES + 255) / 256, 256, 0, stream>>>(deg);
  k_norm<<<(N_EDGES + 255) / 256, 256, 0, stream>>>(ei, ew, deg, norm);

  // T0 = tanh(x @ W_in + b_in)
  k_gemm_in<<<(N_NODES / 16) * (OUT_C / 16) / 8, 256, 0, stream>>>(x, W_in, b_in, T0);

  // T1 = L_hat T0 ; T2 = 2 L_hat T1 - T0   (loop_w == 0)
  k_prop<<<N_EDGES / 8, 256, 0, stream>>>(ei, norm, T0, T1);
  k_prop<<<N_EDGES / 8, 256, 0, stream>>>(ei, norm, T1, T2);
  k_t2<<<(NF + 255) / 256, 256, 0, stream>>>(T2, T0, NF);

  // ACC = relu(sum_k Tk@theta_k + cheb_bias); Y = ACC @ W_out + b_out
  k_gemm_cheb<<<(N_NODES / 16) * (OUT_C / 16) / 8, 256, 0, stream>>>(T0, T1, T2, theta, cb, ACC);
  k_gemm_out<<<((N_NODES / 16) * 3 + 7) / 8, 256, 0, stream>>>(ACC, W_out, b_out, Y);
}